// CDLoss_58643483459894
// MI455X (gfx1250) — compile-verified
//
#include <hip/hip_runtime.h>

typedef __attribute__((ext_vector_type(2))) float v2f;
typedef __attribute__((ext_vector_type(8))) float v8f;

#define B_CLOUDS 16
#define NPTS 4096
#define WAVES 8                         // waves per block (256 threads)
#define BLK (WAVES * 32)
#define CHUNK 2048                      // y points staged in LDS per pass (32 KB)
#define ROWBLOCKS (NPTS / (16 * WAVES)) // 32 row-blocks per cloud
#define TILES_PER_CHUNK (CHUNK / 16)    // 128 column tiles per chunk

// One wave owns a 16-row tile of one cloud; sweeps all 4096 columns.
// Per 16x16 tile: D = A x B + 0 with A=[x | 1], B=[-2y ; y^2] gives
// D[m][n] = -2 x_m.y_n + |y_n|^2. The per-row constant |x_m|^2 cannot
// change argmin_j, so it is added once per row after the min sweep.
__global__ __launch_bounds__(BLK) void chamfer_dir_kernel(
    const float* __restrict__ pred, const float* __restrict__ target,
    float* __restrict__ ws)
{
    __shared__ float4 sy[CHUNK];

    const int dir  = blockIdx.y;             // 0: pred->target, 1: target->pred
    const float* xa = dir ? target : pred;   // row points
    const float* yb = dir ? pred   : target; // column points

    const int blk   = blockIdx.x;
    const int b     = blk / ROWBLOCKS;       // cloud index
    const int rb    = blk % ROWBLOCKS;       // row-block within cloud
    const int tid   = threadIdx.x;
    const int wave  = tid >> 5;
    const int lane  = tid & 31;
    const int l15   = lane & 15;
    const int half  = lane >> 4;

    // ---- A fragment (16x4 f32): lanes 0-15 hold K=0,1; lanes 16-31 hold K=2,3
    const int row = b * NPTS + (rb * WAVES + wave) * 16 + l15;
    const float x0 = xa[row * 3 + 0];
    const float x1 = xa[row * 3 + 1];
    const float x2 = xa[row * 3 + 2];
    const float xs = x0 * x0 + x1 * x1 + x2 * x2;   // added back post-min
    v2f afrag;
    afrag.x = half ? x2   : x0;   // K2 : K0
    afrag.y = half ? 1.0f : x1;   // K3 : K1   (A.K3 = 1 pairs with B.K3 = |y|^2)

    // Per-lane LDS byte offset: lanes 0-15 read K0,K1 half; 16-31 read K2,K3 half
    const int loff = l15 * 16 + half * 8;

    v8f acc;
#pragma unroll
    for (int i = 0; i < 8; ++i) acc[i] = 3.0e38f;

    const v8f czero = {};   // SRC2 inline constant 0

    for (int ch = 0; ch < NPTS / CHUNK; ++ch) {
        __syncthreads();
        // Stage CHUNK column points as float4(-2y0, -2y1, -2y2, |y|^2)
        for (int k = 0; k < CHUNK / BLK; ++k) {
            const int p  = k * BLK + tid;
            const int gp = b * NPTS + ch * CHUNK + p;
            const float y0 = yb[gp * 3 + 0];
            const float y1 = yb[gp * 3 + 1];
            const float y2 = yb[gp * 3 + 2];
            sy[p] = make_float4(-2.f * y0, -2.f * y1, -2.f * y2,
                                y0 * y0 + y1 * y1 + y2 * y2);
        }
        __syncthreads();

        const char* syb = (const char*)sy;
        for (int jt = 0; jt < TILES_PER_CHUNK; jt += 2) {
            // B fragment: direct 8B LDS load, no VALU assembly, conflict-free
            const v2f b0 = *(const v2f*)(syb + jt * 256 + loff);
            const v2f b1 = *(const v2f*)(syb + jt * 256 + 256 + loff);

            const v8f d0 = __builtin_amdgcn_wmma_f32_16x16x4_f32(
                false, afrag, false, b0, (short)0, czero, false, false);
            const v8f d1 = __builtin_amdgcn_wmma_f32_16x16x4_f32(
                false, afrag, false, b1, (short)0, czero, false, false);

#pragma unroll
            for (int i = 0; i < 8; ++i)
                acc[i] = fminf(fminf(acc[i], d0[i]), d1[i]);  // -> v_min3_f32
        }
    }

    // Row-min reduction: butterfly across each 16-lane half (wave32 shuffles)
    float s = 0.f;
#pragma unroll
    for (int i = 0; i < 8; ++i) {
        float r = acc[i];
        r = fminf(r, __shfl_xor(r, 1));
        r = fminf(r, __shfl_xor(r, 2));
        r = fminf(r, __shfl_xor(r, 4));
        r = fminf(r, __shfl_xor(r, 8));
        s += r;   // lanes 0-15: rows 0..7 partial; lanes 16-31: rows 8..15
    }
    s += __shfl_xor(s, 16);   // sum of 16 row-mins (missing +|x|^2 per row)

    // Add back sum of |x_m|^2 over the 16 rows of this tile
    float t = xs;
    t += __shfl_xor(t, 1);
    t += __shfl_xor(t, 2);
    t += __shfl_xor(t, 4);
    t += __shfl_xor(t, 8);    // every lane: sum over rows 0..15
    s += t;

    if (lane == 0) {
        const int idx = (dir * gridDim.x + blk) * WAVES + wave;
        ws[idx] = s;
    }
}

// Deterministic final reduction: fixed strided partials + fixed tree.
__global__ __launch_bounds__(256) void chamfer_reduce_kernel(
    const float* __restrict__ ws, float* __restrict__ out, int n, float scale)
{
    __shared__ float sm[256];
    const int tid = threadIdx.x;
    float s = 0.f;
    for (int i = tid; i < n; i += 256) s += ws[i];
    sm[tid] = s;
    __syncthreads();
    for (int off = 128; off > 0; off >>= 1) {
        if (tid < off) sm[tid] += sm[tid + off];
        __syncthreads();
    }
    if (tid == 0) out[0] = sm[0] * scale;
}

extern "C" void kernel_launch(void* const* d_in, const int* in_sizes, int n_in,
                              void* d_out, int out_size, void* d_ws, size_t ws_size,
                              hipStream_t stream) {
    const float* pred   = (const float*)d_in[0];
    const float* target = (const float*)d_in[1];
    // d_in[2] (batch) is repeat(arange(B), N) by construction -> dense layout
    // is exactly row-major [B, N, 3]; no scatter needed.
    float* ws  = (float*)d_ws;
    float* out = (float*)d_out;

    dim3 grid(B_CLOUDS * ROWBLOCKS, 2);   // 512 blocks x 2 directions
    chamfer_dir_kernel<<<grid, BLK, 0, stream>>>(pred, target, ws);

    const int npart = 2 * B_CLOUDS * ROWBLOCKS * WAVES;   // 8192 partials
    chamfer_reduce_kernel<<<1, 256, 0, stream>>>(
        ws, out, npart, 1.0f / (float)(B_CLOUDS * NPTS));
}